// CALayer_32993938768167
// MI455X (gfx1250) — compile-verified
//
#include <hip/hip_runtime.h>
#include <cstddef>

typedef float v2f __attribute__((ext_vector_type(2)));
typedef float v8f __attribute__((ext_vector_type(8)));

#define Bn 8
#define Cc 64
#define Hh 128
#define Ww 128
#define HW (Hh * Ww)

#if __has_builtin(__builtin_amdgcn_global_load_async_to_lds_b32) && \
    __has_builtin(__builtin_amdgcn_s_wait_asynccnt)
#define USE_ASYNC_LDS 1
#else
#define USE_ASYNC_LDS 0
#endif

// One element global -> LDS, via CDNA5 async DMA (ASYNCcnt) when available.
__device__ inline void lds_copy1(const float* g, float* l) {
#if USE_ASYNC_LDS
  __builtin_amdgcn_global_load_async_to_lds_b32(
      (__attribute__((address_space(1))) int*)g,
      (__attribute__((address_space(3))) int*)l, 0, 0);
#else
  *l = *g;
#endif
}

__device__ inline void lds_fence_barrier() {
#if USE_ASYNC_LDS
  __builtin_amdgcn_s_wait_asynccnt(0);
#endif
  __syncthreads();
}

// D = A(16x4) * B(4x16) + C stepped over K by 4 (fp32 WMMA).
// A rows are pre-staged contiguously in LDS: fragment = one ds_load_b64.
// Layout (CDNA5 ISA 7.12.2): a[v], lane L: m = L&15, k = kb + v + 2*(L>>4);
// B symmetric with n = L&15. C/D: vgpr r, lane L -> (m = r + 8*(L>>4), n = L&15).
template <int K, typename FB>
__device__ inline v8f gemm_lA(int lane, const float* Arow, FB&& loadB,
                              v8f acc) {
  const int mn = lane & 15;
  const int hi2 = (lane >> 4) << 1; // 0 or 2
#pragma unroll 4
  for (int kb = 0; kb < K; kb += 4) {
    v2f a = *(const v2f*)(Arow + kb + hi2); // ds_load_b64, 8B aligned
    v2f b;
    b.x = loadB(kb + hi2 + 0, mn);
    b.y = loadB(kb + hi2 + 1, mn);
    acc = __builtin_amdgcn_wmma_f32_16x16x4_f32(false, a, false, b,
                                                (short)0, acc, false, false);
  }
  return acc;
}

// Stage rows h-1..h+1 of a [64][H][W] tensor into LDS [64][3][130] with zero
// halo columns. No fence here; caller runs lds_fence_barrier().
__device__ inline void stage_rows3(const float* __restrict__ src,
                                   float* __restrict__ buf, int h) {
  for (int i = threadIdx.x; i < Cc * 390; i += blockDim.x) {
    int c = i / 390, rem = i % 390, r = rem / 130, ci = rem % 130;
    int hh = h - 1 + r, xc = ci - 1;
    if (hh >= 0 && hh < Hh && xc >= 0 && xc < Ww)
      lds_copy1(src + c * HW + hh * Ww + xc, buf + i);
    else
      buf[i] = 0.f;
  }
}

// ---------------------------------------------------------------------------
// K1: fused front end. One block = one (b, h). 256 threads = 8 wave32,
// each wave owns 16 pixel columns. Produces w_pre[B,72,H,W] and xv[B,64,H,W].
// ---------------------------------------------------------------------------
__global__ void k1_frontend(const float* __restrict__ x,
                            const float* __restrict__ wkey,  // [64,8,3,3]
                            const float* __restrict__ we1,   // [32,64]
                            const float* __restrict__ we2,   // [72,16]
                            const float* __restrict__ be2,   // [72]
                            const float* __restrict__ wc1,   // [64,32]
                            float* __restrict__ wpre,        // [B,72,H,W]
                            float* __restrict__ xv) {        // [B,64,H,W]
  extern __shared__ float smem[];
  float* xbuf  = smem;                   // [64][3][130]          24960 f
  float* wkL   = xbuf + 64 * 3 * 130;    // [4][16][144] padded    9216 f
  float* we1L  = wkL + 9216;             // [32][64]               2048 f
  float* we2L  = we1L + 2048;            // [2][48][16] padded     1536 f
  float* wc1L  = we2L + 1536;            // [64][32]               2048 f
  float* ktile = wc1L + 2048;            // [8 waves][64][16]      8192 f
  float* h1t   = ktile + 8192;           // [8 waves][32][16]      4096 f

  const int bh = blockIdx.x;
  const int b = bh / Hh, h = bh % Hh;
  const int tid = threadIdx.x;
  const int lane = tid & 31;
  const int wave = tid >> 5;
  const int p0 = wave * 16;
  const int mn = lane & 15;
  const int hi = lane >> 4;
  const int col = p0 + mn;

  // ---- stage x halo rows + all weights (pre-padded to WMMA A layout).
  stage_rows3(x + (size_t)b * Cc * HW, xbuf, h);
  for (int i = tid; i < 4 * 16 * 144; i += blockDim.x) {
    int pair = i / 2304, rm = i % 2304, m = rm / 144, k = rm % 144;
    int kk = k % 72;
    if ((k < 72) == (m < 8))
      lds_copy1(wkey + ((16 * pair + m) * 8 + kk / 9) * 9 + (kk % 9), wkL + i);
    else
      wkL[i] = 0.f;
  }
  for (int i = tid; i < 2048; i += blockDim.x) lds_copy1(we1 + i, we1L + i);
  for (int i = tid; i < 2 * 48 * 16; i += blockDim.x) {
    int g = i / 768, rm = i % 768, m = rm / 16, k = rm % 16;
    if (m < 36)
      lds_copy1(we2 + (36 * g + m) * 16 + k, we2L + i);
    else
      we2L[i] = 0.f;
  }
  for (int i = tid; i < 2048; i += blockDim.x) lds_copy1(wc1 + i, wc1L + i);
  lds_fence_barrier();

  // ---- key conv: grouped 3x3 (8 groups x 8ic x 8oc) as implicit GEMM.
  // Two groups stacked per tile: M=16 (oc 16t..16t+15), block-diagonal K=144.
  for (int t = 0; t < 4; ++t) {
    v8f acc = {};
    acc = gemm_lA<144>(lane, wkL + t * 2304 + mn * 144,
        [&](int k, int nn) -> float {
          int kk = k % 72;
          int ic = 16 * t + ((k >= 72) ? 8 : 0) + kk / 9;
          int tap = kk % 9;
          return xbuf[ic * 390 + (tap / 3) * 130 + (p0 + nn + tap % 3)];
        },
        acc);
#pragma unroll
    for (int r = 0; r < 8; ++r) {
      int ch = 16 * t + r + 8 * hi;
      ktile[(wave * 64 + ch) * 16 + mn] = fmaxf(acc[r], 0.f); // ReLU
    }
  }
  // LDS is in-order per wave: ktile reads below are safe without a barrier.

  // ---- e1: 1x1 conv, 2 groups, per group out[16] = W[16x64] * qk_g[64].
  // qk channel z: even -> x[32g + z/2], odd -> k[32g + z/2].
  for (int ge = 0; ge < 2; ++ge) {
    v8f acc = {};
    acc = gemm_lA<64>(lane, we1L + (16 * ge + mn) * 64,
        [&](int k, int nn) -> float {
          int c = 32 * ge + (k >> 1);
          return (k & 1) ? ktile[(wave * 64 + c) * 16 + nn]
                         : xbuf[c * 390 + 130 + (p0 + nn + 1)];
        },
        acc);
#pragma unroll
    for (int r = 0; r < 8; ++r) {
      int ch = 16 * ge + r + 8 * hi;
      h1t[(wave * 32 + ch) * 16 + mn] = fmaxf(acc[r], 0.f); // ReLU
    }
  }

  // ---- e2: 1x1 conv, 2 groups, out[36] = W[36x16] * h1_g[16] (+bias).
  float* wpb = wpre + (size_t)b * 72 * HW + (size_t)h * Ww;
  for (int g2 = 0; g2 < 2; ++g2) {
    for (int mb = 0; mb < 48; mb += 16) {
      v8f acc = {};
      acc = gemm_lA<16>(lane, we2L + g2 * 768 + (mb + mn) * 16,
          [&](int k, int nn) -> float {
            return h1t[(wave * 32 + 16 * g2 + k) * 16 + nn];
          },
          acc);
#pragma unroll
      for (int r = 0; r < 8; ++r) {
        int mi = mb + r + 8 * hi;
        if (mi < 36) {
          int ch72 = 36 * g2 + mi;
          wpb[(size_t)ch72 * HW + col] = acc[r] + be2[ch72];
        }
      }
    }
  }

  // ---- c1 (value path): 1x1 conv, 2 groups, out[32] = W[32x32] * x_g[32].
  float* xvb = xv + (size_t)b * 64 * HW + (size_t)h * Ww;
  for (int gg = 0; gg < 2; ++gg) {
    for (int mb = 0; mb < 32; mb += 16) {
      v8f acc = {};
      acc = gemm_lA<32>(lane, wc1L + (32 * gg + mb + mn) * 32,
          [&](int k, int nn) -> float {
            return xbuf[(32 * gg + k) * 390 + 130 + (p0 + nn + 1)];
          },
          acc);
#pragma unroll
      for (int r = 0; r < 8; ++r) {
        int oc = 32 * gg + mb + r + 8 * hi;
        xvb[(size_t)oc * HW + col] = acc[r];
      }
    }
  }
}

// ---------------------------------------------------------------------------
// K2: GroupNorm statistics. 64 blocks = (b, group). Emits mean and rstd.
// ---------------------------------------------------------------------------
__global__ void k2_gnstats(const float* __restrict__ wpre,
                           float* __restrict__ stats) {
  __shared__ float ss[256], sq[256];
  int bg = blockIdx.x;
  int b = bg >> 3, g = bg & 7;
  const float* base = wpre + (size_t)b * 72 * HW + (size_t)g * 9 * HW;
  float s = 0.f, q = 0.f;
  for (int i = threadIdx.x; i < 9 * HW; i += blockDim.x) {
    float v = base[i];
    s += v;
    q += v * v;
  }
  ss[threadIdx.x] = s;
  sq[threadIdx.x] = q;
  __syncthreads();
  for (int o = 128; o > 0; o >>= 1) {
    if (threadIdx.x < o) {
      ss[threadIdx.x] += ss[threadIdx.x + o];
      sq[threadIdx.x] += sq[threadIdx.x + o];
    }
    __syncthreads();
  }
  if (threadIdx.x == 0) {
    float N = 9.f * HW;
    float mean = ss[0] / N;
    float var = sq[0] / N - mean * mean;
    stats[bg * 2] = mean;
    stats[bg * 2 + 1] = rsqrtf(var + 1e-5f);
  }
}

// ---------------------------------------------------------------------------
// K3: normalize dynamic kernels + 3x3 unfold + per-pixel local conv.
// One block = (b, h). Thread t: pixel t&127, channel-half t>>7.
// ---------------------------------------------------------------------------
__global__ void k3_localconv(const float* __restrict__ xv,
                             const float* __restrict__ wpre,
                             const float* __restrict__ stats,
                             const float* __restrict__ gnw,
                             const float* __restrict__ gnb,
                             float* __restrict__ out) {
  extern __shared__ float xvbuf[]; // [64][3][130]
  int bh = blockIdx.x;
  int b = bh / Hh, h = bh % Hh;
  int tid = threadIdx.x;

  stage_rows3(xv + (size_t)b * 64 * HW, xvbuf, h);
  lds_fence_barrier();

  int px = tid & 127;
  int half = tid >> 7; // b2 = 2b + half, channels half*32..half*32+31
  // Normalized per-pixel kernels: 4 share-groups x 9 taps.
  float wn[4][9];
  const float* wpb = wpre + (size_t)b * 72 * HW + (size_t)half * 36 * HW +
                     (size_t)h * Ww + px;
  for (int g = 0; g < 4; ++g) {
    int gr = half * 4 + g; // GN group = ch72/9
    float mean = stats[(b * 8 + gr) * 2];
    float rstd = stats[(b * 8 + gr) * 2 + 1];
#pragma unroll
    for (int t = 0; t < 9; ++t) {
      int ch72 = half * 36 + g * 9 + t;
      float v = wpb[(size_t)(g * 9 + t) * HW];
      wn[g][t] = (v - mean) * rstd * gnw[ch72] + gnb[ch72];
    }
  }
  float* ob = out + (size_t)b * 64 * HW + (size_t)half * 32 * HW +
              (size_t)h * Ww + px;
  for (int og = 0; og < 32; ++og) {
    int g = og >> 3; // share_planes = 8
    int c = half * 32 + og;
    float acc = 0.f;
#pragma unroll
    for (int t = 0; t < 9; ++t) {
      acc += xvbuf[c * 390 + (t / 3) * 130 + (px + t % 3)] * wn[g][t];
    }
    ob[(size_t)og * HW] = acc;
  }
}

// ---------------------------------------------------------------------------
// K4: per-(b,c) spatial mean of pre-attention output. 512 blocks.
// ---------------------------------------------------------------------------
__global__ void k4_camean(const float* __restrict__ out,
                          float* __restrict__ cam) {
  __shared__ float ss[256];
  int bc = blockIdx.x;
  const float* base = out + (size_t)bc * HW;
  float s = 0.f;
  for (int i = threadIdx.x; i < HW; i += blockDim.x) s += base[i];
  ss[threadIdx.x] = s;
  __syncthreads();
  for (int o = 128; o > 0; o >>= 1) {
    if (threadIdx.x < o) ss[threadIdx.x] += ss[threadIdx.x + o];
    __syncthreads();
  }
  if (threadIdx.x == 0) cam[bc] = ss[0] / (float)HW;
}

// ---------------------------------------------------------------------------
// K5: channel attention MLP (64 -> 4 relu -> 64 sigmoid). 8 blocks x 64 thr.
// ---------------------------------------------------------------------------
__global__ void k5_attn(const float* __restrict__ cam,
                        const float* __restrict__ wdu1, // [4,64]
                        const float* __restrict__ wdu2, // [64,4]
                        float* __restrict__ y) {
  __shared__ float m[64], y1[4];
  int b = blockIdx.x;
  int t = threadIdx.x;
  m[t] = cam[b * 64 + t];
  __syncthreads();
  if (t < 4) {
    float s = 0.f;
    for (int c = 0; c < 64; ++c) s += m[c] * wdu1[t * 64 + c];
    y1[t] = fmaxf(s, 0.f);
  }
  __syncthreads();
  float s = 0.f;
#pragma unroll
  for (int j = 0; j < 4; ++j) s += y1[j] * wdu2[t * 4 + j];
  y[b * 64 + t] = 1.f / (1.f + __expf(-s));
}

// ---------------------------------------------------------------------------
// K6: scale output in place by per-(b,c) attention.
// ---------------------------------------------------------------------------
__global__ void k6_scale(float* __restrict__ out, const float* __restrict__ y) {
  size_t total = (size_t)Bn * 64 * HW;
  for (size_t i = (size_t)blockIdx.x * blockDim.x + threadIdx.x; i < total;
       i += (size_t)gridDim.x * blockDim.x) {
    out[i] *= y[i / HW];
  }
}

extern "C" void kernel_launch(void* const* d_in, const int* in_sizes, int n_in,
                              void* d_out, int out_size, void* d_ws,
                              size_t ws_size, hipStream_t stream) {
  const float* x    = (const float*)d_in[0];
  const float* wkey = (const float*)d_in[1];
  const float* we1  = (const float*)d_in[2];
  const float* we2  = (const float*)d_in[3];
  const float* be2  = (const float*)d_in[4];
  const float* gnw  = (const float*)d_in[5];
  const float* gnb  = (const float*)d_in[6];
  const float* wc1  = (const float*)d_in[7];
  const float* wdu1 = (const float*)d_in[8];
  const float* wdu2 = (const float*)d_in[9];
  float* out = (float*)d_out;

  float* ws    = (float*)d_ws;
  float* wpre  = ws;                            // B*72*HW = 9,437,184 floats
  float* xvbuf = wpre + (size_t)Bn * 72 * HW;   // B*64*HW = 8,388,608
  float* stats = xvbuf + (size_t)Bn * 64 * HW;  // 128
  float* cam   = stats + 128;                   // 512
  float* yv    = cam + 512;                     // 512

  size_t smem1 = (size_t)(64 * 3 * 130 + 9216 + 2048 + 1536 + 2048 +
                          8 * 64 * 16 + 8 * 32 * 16) *
                 sizeof(float); // ~208 KB, fits 320 KB WGP LDS
  size_t smem3 = (size_t)(64 * 3 * 130) * sizeof(float); // ~100 KB

  k1_frontend<<<Bn * Hh, 256, smem1, stream>>>(x, wkey, we1, we2, be2, wc1,
                                               wpre, xvbuf);
  k2_gnstats<<<64, 256, 0, stream>>>(wpre, stats);
  k3_localconv<<<Bn * Hh, 256, smem3, stream>>>(xvbuf, wpre, stats, gnw, gnb,
                                                out);
  k4_camean<<<Bn * 64, 256, 0, stream>>>(out, cam);
  k5_attn<<<Bn, 64, 0, stream>>>(cam, wdu1, wdu2, yv);
  k6_scale<<<2048, 256, 0, stream>>>(out, yv);
}